// OptimizedGRUModule_49933289783562
// MI455X (gfx1250) — compile-verified
//
#include <hip/hip_runtime.h>

typedef __attribute__((ext_vector_type(2))) float v2f;
typedef __attribute__((ext_vector_type(8))) float v8f;

#define F_TOT 65   // per-row: [h, x0..x63]
#define F_IN  64

// One wave = 32 rows (two 16-row WMMA tiles).
// D = A(16x4: W_ih, M=gate) x B(4x16: x^T, N=row), accumulated over 16 K-blocks.
// f32 WMMA keeps full reference precision; kernel is HBM-bandwidth bound
// (~138 MB traffic -> ~6 us floor at 23.3 TB/s; compute ~0.2 GFLOP is free).
// Each lane forms ONE pointer per tile (lane-dependent part folded in), then
// all 16 K-block loads use compile-time-constant offsets so they vectorize to
// global_load_b64 with immediate offsets (no per-load address VALU).
__global__ __launch_bounds__(256) void gru_wmma_f32_kernel(
    const float* __restrict__ inp,   // [rows, 65]
    const float* __restrict__ W_ih,  // [3, 64]
    const float* __restrict__ W_hh,  // [3, 1]
    const float* __restrict__ b_ih,  // [3]
    const float* __restrict__ b_hh,  // [3]
    const float* __restrict__ w_out, // [1,1]
    const float* __restrict__ b_out, // [1]
    float* __restrict__ out,         // [rows]
    int rows)
{
    const unsigned lane = threadIdx.x & 31u;
    const unsigned n    = lane & 15u;   // row-in-tile (B/D) or gate index (A)
    const unsigned hi   = lane >> 4;    // half-wave: K pair {0,1} vs {2,3}
    const unsigned wave = threadIdx.x >> 5;
    const unsigned base = (blockIdx.x * 8u + wave) * 32u;
    if (base + 32u > (unsigned)rows) return;  // whole-wave guard (rows%32==0)

    // ---- A fragments: W_ih with gate=M (rows 3..15 zero), per K-block ----
    // A layout (16x4 f32): v0 = A[m][2*hi], v1 = A[m][2*hi+1], m = lane&15
    v2f aw[16];
    #pragma unroll
    for (int kb = 0; kb < 16; ++kb) { aw[kb].x = 0.0f; aw[kb].y = 0.0f; }
    if (n < 3u) {
        const float* wp = W_ih + (n * F_IN + 2u * hi);  // one lane pointer
        #pragma unroll
        for (int kb = 0; kb < 16; ++kb) {               // constant offsets
            aw[kb].x = wp[4 * kb];
            aw[kb].y = wp[4 * kb + 1];
        }
    }

    // ---- B fragments: x^T for two row-tiles; one pointer per tile ----
    const float* p0 = inp + ((size_t)(base + n) * F_TOT + 1u + 2u * hi);
    const float* p1 = p0 + (size_t)16 * F_TOT;

    v8f c0 = {};  // D[gate][row] tile 0
    v8f c1 = {};  // D[gate][row] tile 1
    #pragma unroll
    for (int kb = 0; kb < 16; ++kb) {
        v2f b0, b1;
        b0.x = p0[4 * kb];
        b0.y = p0[4 * kb + 1];
        b1.x = p1[4 * kb];
        b1.y = p1[4 * kb + 1];
        c0 = __builtin_amdgcn_wmma_f32_16x16x4_f32(
                 false, aw[kb], false, b0, (short)0, c0, false, false);
        c1 = __builtin_amdgcn_wmma_f32_16x16x4_f32(
                 false, aw[kb], false, b1, (short)0, c1, false, false);
    }

    // ---- Epilogue: lane L handles row base+L (lanes 16-31 take tile 1) ----
    // D layout: c[j], lanes 0-15 hold M=j, N=lane -> gi[gate j][row n].
    float g0 = c0[0], g1 = c0[1], g2 = c0[2];
    const float t0 = __shfl(c1[0], (int)n, 32);
    const float t1 = __shfl(c1[1], (int)n, 32);
    const float t2 = __shfl(c1[2], (int)n, 32);
    if (hi) { g0 = t0; g1 = t1; g2 = t2; }

    const unsigned row = base + lane;
    const float h = inp[(size_t)row * F_TOT];  // initial hidden state

    const float gr  = g0 + b_ih[0] + h * W_hh[0] + b_hh[0];
    const float gz  = g1 + b_ih[1] + h * W_hh[1] + b_hh[1];
    const float ghn =                h * W_hh[2] + b_hh[2];

    const float r  = 1.0f / (1.0f + __expf(-gr));
    const float z  = 1.0f / (1.0f + __expf(-gz));
    const float gn = g2 + b_ih[2] + r * ghn;
    const float e2 = __expf(2.0f * gn);
    const float nn = 1.0f - 2.0f / (e2 + 1.0f);   // tanh, overflow-safe
    const float hn = (1.0f - z) * nn + z * h;

    out[row] = hn * w_out[0] + b_out[0];
}

extern "C" void kernel_launch(void* const* d_in, const int* in_sizes, int n_in,
                              void* d_out, int out_size, void* d_ws, size_t ws_size,
                              hipStream_t stream) {
    const float* inp   = (const float*)d_in[0];
    const float* W_ih  = (const float*)d_in[1];
    const float* W_hh  = (const float*)d_in[2];
    const float* b_ih  = (const float*)d_in[3];
    const float* b_hh  = (const float*)d_in[4];
    const float* w_out = (const float*)d_in[5];
    const float* b_out = (const float*)d_in[6];
    float* out = (float*)d_out;

    const int rows = in_sizes[0] / F_TOT;          // 524288
    const int blocks = (rows + 255) / 256;         // 256 thr = 8 waves * 32 rows
    gru_wmma_f32_kernel<<<blocks, 256, 0, stream>>>(
        inp, W_ih, W_hh, b_ih, b_hh, w_out, b_out, out, rows);
}